// SelectDropMAX_32040456028967
// MI455X (gfx1250) — compile-verified
//
#include <hip/hip_runtime.h>
#include <stdint.h>

// SelectDropMAX: x[B,C,H,W] -> x - 0.9*x*(peak | random_block)
// B=64 C=512 H=W=28  -> 32768 channels x 784 floats (3136 B each).
// Memory-bound (~205 MB total traffic, ~8.8 us at 23.3 TB/s).
// One wave32 per channel; channel staged to LDS via gfx1250 async
// global->LDS B128 copies (GVS: scalar base + one lane-offset VGPR,
// per-issue immediate offset applied to BOTH LDS and global address),
// tile pulled into registers once, max-reduced, masked, stored B128.

#define HW        784
#define TILE_F4   196          // 784 floats = 196 float4
#define PAD_F     896          // padded tile: 3584 B = 7*512 B per wave
#define WAVES_PB  8
#define THREADS   (WAVES_PB * 32)
#define NCHAN     (64 * 512)

// One async B128 issue: INST_OFFSET is added to both LDS dest and global src.
#define ASYNC_LD(OFF)                                                        \
    asm volatile("global_load_async_to_lds_b128 %0, %1, %2 offset:" #OFF     \
                 :: "v"(laddr), "v"(goff), "s"(sbase) : "memory")

__global__ __launch_bounds__(THREADS)
void SelectDropMAX_kernel(const float* __restrict__ x,
                          const int*   __restrict__ mask_y,
                          const int*   __restrict__ mask_x,
                          float*       __restrict__ out)
{
    __shared__ float lds[WAVES_PB * PAD_F];   // 28672 B of the 320 KB WGP LDS

    const int lane = threadIdx.x & 31;
    const int wave = threadIdx.x >> 5;
    const int ch   = blockIdx.x * WAVES_PB + wave;   // exact: 4096*8 = 32768

    const float* src  = x   + (size_t)ch * HW;
    float*       dst  = out + (size_t)ch * HW;
    float*       tile = lds + wave * PAD_F;

    // Wave-uniform scalar base for GVS async loads.
    const uint64_t gb  = (uint64_t)(uintptr_t)src;
    const uint32_t blo = __builtin_amdgcn_readfirstlane((uint32_t)gb);
    const uint32_t bhi = __builtin_amdgcn_readfirstlane((uint32_t)(gb >> 32));
    const uint64_t sbase = ((uint64_t)bhi << 32) | blo;

    // ---- Stage 3136 B global -> LDS: 6 full B128 issues sharing one
    // offset/address VGPR pair (immediate offset:512*i covers both sides),
    // plus one clamped tail issue (lanes >= 4 duplicate the last valid 16B
    // chunk into the LDS pad region -> harmless, global read stays in-bounds).
    const uint32_t lds_base = (uint32_t)(uintptr_t)tile;  // low 32b = LDS offset
    {
        uint32_t goff  = (uint32_t)(lane * 16);
        uint32_t laddr = lds_base + goff;
        ASYNC_LD(0);
        ASYNC_LD(512);
        ASYNC_LD(1024);
        ASYNC_LD(1536);
        ASYNC_LD(2048);
        ASYNC_LD(2560);
        // Tail: bytes 3072..3135 valid (lanes 0..3); clamp global, pad LDS.
        uint32_t tb    = 3072u + (uint32_t)(lane * 16);
        uint32_t tgoff = tb > 3120u ? 3120u : tb;
        uint32_t tladdr = lds_base + tb;
        asm volatile("global_load_async_to_lds_b128 %0, %1, %2"
                     :: "v"(tladdr), "v"(tgoff), "s"(sbase) : "memory");
    }

    // Overlap the mask-corner loads with the async copy.
    const int y0 = mask_y[ch];
    const int x0 = mask_x[ch];

    asm volatile("s_wait_asynccnt 0x0" ::: "memory");

    // ---- Pass 1: pull tile into registers (LDS read once), per-lane max,
    //      then wave32 butterfly reduction.
    const float4* tp = (const float4*)tile;
    float4 vreg[7];
    float  m = -__builtin_inff();
#pragma unroll
    for (int i = 0; i < 7; ++i) {
        int vidx = lane + 32 * i;                 // iter 6: lanes 0..3 valid
        if (vidx < TILE_F4) {
            float4 v = tp[vidx];
            vreg[i]  = v;
            m = fmaxf(m, fmaxf(fmaxf(v.x, v.y), fmaxf(v.z, v.w)));
        }
    }
#pragma unroll
    for (int off = 16; off > 0; off >>= 1)
        m = fmaxf(m, __shfl_xor(m, off, 32));

    // ---- Pass 2: apply mask from registers, coalesced B128 stores.
#pragma unroll
    for (int i = 0; i < 7; ++i) {
        int vidx = lane + 32 * i;
        if (vidx < TILE_F4) {
            float4 v    = vreg[i];
            float  e[4] = {v.x, v.y, v.z, v.w};
            int    base = vidx << 2;
            float  o[4];
#pragma unroll
            for (int j = 0; j < 4; ++j) {
                int  idx  = base + j;
                int  h    = idx / 28;
                int  w    = idx - h * 28;
                bool rnd  = (h >= y0) & (h < y0 + 7) & (w >= x0) & (w < x0 + 2);
                bool peak = (e[j] == m);
                float t   = 0.9f * e[j];          // mul then sub, matches ref
                o[j]      = (rnd | peak) ? (e[j] - t) : e[j];
            }
            float4 ov = {o[0], o[1], o[2], o[3]};
            ((float4*)dst)[vidx] = ov;
        }
    }
}

extern "C" void kernel_launch(void* const* d_in, const int* in_sizes, int n_in,
                              void* d_out, int out_size, void* d_ws, size_t ws_size,
                              hipStream_t stream) {
    (void)in_sizes; (void)n_in; (void)out_size; (void)d_ws; (void)ws_size;
    const float* x  = (const float*)d_in[0];
    const int*   my = (const int*)d_in[1];
    const int*   mx = (const int*)d_in[2];
    float*       o  = (float*)d_out;
    dim3 grid(NCHAN / WAVES_PB);   // 4096 blocks x 256 threads (8 wave32)
    dim3 block(THREADS);
    SelectDropMAX_kernel<<<grid, block, 0, stream>>>(x, my, mx, o);
}